// TerrainInterpolator_63926293234154
// MI455X (gfx1250) — compile-verified
//
#include <hip/hip_runtime.h>
#include <hip/hip_bf16.h>
#include <stdint.h>

#define NXG 4096
#define NYG 4096
#define THREADS 256   // 8 wave32 waves per block
#define BLOCKS  2048  // ~4.2M points / 524288 threads -> 8 grid-stride iters

// Branchless lower_bound over an LDS-resident sorted array of 4096 floats.
// Returns #elements < v, i.e. searchsorted(side='left'), with the property
// that the result is already clamped to <= 4095 (steps sum to 4095 and the
// max probe index is 4094, so no out-of-bounds LDS read).
__device__ __forceinline__ int lower_bound12(const float* __restrict__ b, float v) {
    int L = 0;
#pragma unroll
    for (int step = 2048; step > 0; step >>= 1) {
        L += (b[L + step - 1] < v) ? step : 0;   // 12 dependent ds_load_b32
    }
    return L;
}

__global__ __launch_bounds__(THREADS)
void TerrainInterpolator_63926293234154_kernel(
    const float* __restrict__ qx, const float* __restrict__ qy,
    const float* __restrict__ xs, const float* __restrict__ ys,
    const float* __restrict__ zs, float* __restrict__ out, int n)
{
    __shared__ __align__(16) float s_xs[NXG];   // 16 KB
    __shared__ __align__(16) float s_ys[NYG];   // 16 KB  (32 KB of 320 KB/WGP)

    const int tid = threadIdx.x;

    // ---- CDNA5 async DMA fill: global -> LDS, tracked by ASYNCcnt ----------
    // 16 KB per array = 1024 x b128 lane-transfers = 4 iters x 256 lanes.
    {
        uint32_t lds_x = (uint32_t)(uintptr_t)&s_xs[0];  // flat LDS addr low 32b == LDS byte offset
        uint32_t lds_y = (uint32_t)(uintptr_t)&s_ys[0];
#pragma unroll
        for (int i = 0; i < 4; ++i) {
            uint32_t off = (uint32_t)(tid + i * THREADS) * 16u;  // byte offset, 16B aligned
            uint32_t lx  = lds_x + off;
            uint32_t ly  = lds_y + off;
            // GVS mode: mem_addr = SGPR64 + VGPR32_offset ; VDST = LDS byte address
            asm volatile("global_load_async_to_lds_b128 %0, %1, %2"
                         :: "v"(lx), "v"(off), "s"(xs) : "memory");
            asm volatile("global_load_async_to_lds_b128 %0, %1, %2"
                         :: "v"(ly), "v"(off), "s"(ys) : "memory");
        }
        asm volatile("s_wait_asynccnt 0" ::: "memory");
    }
    __syncthreads();

    // ---- grid-stride query loop -------------------------------------------
    const int stride = (int)(gridDim.x * blockDim.x);
    for (int p = (int)(blockIdx.x * blockDim.x) + tid; p < n; p += stride) {
        const int pn = p + stride;
        if (pn < n) {                       // prefetch next iteration's queries
            __builtin_prefetch(&qx[pn], 0, 0);   // -> global_prefetch_b8
            __builtin_prefetch(&qy[pn], 0, 0);
        }

        // Streaming, use-once data: non-temporal so zs stays hot in L2.
        const float px = __builtin_nontemporal_load(&qx[p]);
        const float py = __builtin_nontemporal_load(&qy[p]);

        // idx_right (clamped) and idx_left per reference semantics
        const int Rx = lower_bound12(s_xs, px);
        const int Ry = lower_bound12(s_ys, py);
        const int Lx = (Rx > 0) ? Rx - 1 : 0;
        const int Ly = (Ry > 0) ? Ry - 1 : 0;

        float dw = fmaxf(px - s_xs[Lx], 0.0f);   // dist_west  (x left)
        float de = fmaxf(s_xs[Rx] - px, 0.0f);   // dist_east  (x right)
        const bool zx = (dw == 0.0f) & (de == 0.0f);
        dw = zx ? 1.0f : dw;
        de = zx ? 1.0f : de;

        float dsn = fmaxf(py - s_ys[Ly], 0.0f);  // dist_south (y left)
        float dn  = fmaxf(s_ys[Ry] - py, 0.0f);  // dist_north (y right)
        const bool zy = (dsn == 0.0f) & (dn == 0.0f);
        dsn = zy ? 1.0f : dsn;
        dn  = zy ? 1.0f : dn;

        // 4 gathers; zs is 64 MB -> fully L2-resident (192 MB L2).
        // Ly/Ry are adjacent columns -> the two loads per row share a line.
        const float* rw = zs + (size_t)Lx * NYG;
        const float* re = zs + (size_t)Rx * NYG;
        const float swv = rw[Ly];
        const float nwv = rw[Ry];
        const float sev = re[Ly];
        const float nev = re[Ry];

        const float num = swv * dn * de + nwv * dsn * de
                        + nev * dsn * dw + sev * dn * dw;
        const float den = (dw + de) * (dn + dsn);
        __builtin_nontemporal_store(num / den, &out[p]);
    }
}

extern "C" void kernel_launch(void* const* d_in, const int* in_sizes, int n_in,
                              void* d_out, int out_size, void* d_ws, size_t ws_size,
                              hipStream_t stream) {
    const float* qx = (const float*)d_in[0];  // interp_xs, 4194304
    const float* qy = (const float*)d_in[1];  // interp_ys, 4194304
    const float* xs = (const float*)d_in[2];  // xs, 4096 sorted
    const float* ys = (const float*)d_in[3];  // ys, 4096 sorted
    const float* zs = (const float*)d_in[4];  // zs, 4096*4096
    float* out = (float*)d_out;
    const int n = out_size;

    TerrainInterpolator_63926293234154_kernel<<<dim3(BLOCKS), dim3(THREADS), 0, stream>>>(
        qx, qy, xs, ys, zs, out, n);
}